// QuantizedBn2d_32590211842402
// MI455X (gfx1250) — compile-verified
//
#include <hip/hip_runtime.h>

// ---------------------------------------------------------------------------
// QuantizedBn2d requant: elementwise int8 requantization, HBM-streaming bound.
// 411 MB traffic @ 23.3 TB/s => ~17.6us floor. Strategy: double-buffered
// async global->LDS pipeline (CDNA5 ASYNCcnt path) so each wave keeps
// 4 x B128 async loads in flight while computing the previous tile from LDS.
// Rolled steady-state loop keeps code small (I$-friendly); NT stores keep the
// write-once output from rinsing L2.
// ---------------------------------------------------------------------------

#define N_ELEMS   51380224   // 64*256*56*56
#define N_VEC4    12845056   // N/4
#define NCHAN     256
#define HW4       784        // (56*56)/4 vec4 per channel plane

#define THREADS   256
#define WAVES     8
#define V4_TILE   1024       // vec4 per tile (16 KB)
#define V4_WAVE   128        // vec4 per wave per tile
#define TILES_PB  8
#define NBLOCKS   1568       // 1568 * 8 * 1024 == N_VEC4 exactly

typedef float f4 __attribute__((ext_vector_type(4)));

#if defined(__gfx1250__) && __has_builtin(__builtin_amdgcn_global_load_async_to_lds_b128)
#define USE_ASYNC 1
#else
#define USE_ASYNC 0
#endif

#if USE_ASYNC
// Builtin expects: (int4 AS(1)*, int4 AS(3)*, imm offset, imm cpol)
typedef int v4i_vec __attribute__((__vector_size__(4 * sizeof(int))));
typedef __attribute__((address_space(1))) v4i_vec *g_v4i_ptr;
typedef __attribute__((address_space(3))) v4i_vec *l_v4i_ptr;
#endif

// Exact gemmlowp: SaturatingRoundingDoublingHighMul + rounding arith rshift.
__device__ __forceinline__ float requant1(int s32, int lsh, int sh,
                                          long long mask, long long half,
                                          int M0, int z3) {
    long long s    = (long long)s32 << lsh;          // shift<0 path pre-shift
    long long prod = s * (long long)M0;
    long long nudg = (prod >= 0) ? (1ll << 30) : (1ll - (1ll << 30));
    long long t    = prod + nudg;
    // truncate-toward-zero divide by 2^31
    long long high = (t >= 0) ? (t >> 31) : -((-t) >> 31);
    long long tot  = high;
    if (sh > 0) {
        long long rem = high & mask;
        long long thr = half + (long long)(high < 0);
        tot = (high >> sh) + (long long)(rem > thr);
    }
    tot += (long long)z3;
    tot = tot < -128ll ? -128ll : (tot > 127ll ? 127ll : tot);
    return (float)tot;
}

__global__ __launch_bounds__(THREADS)
void requant_bn_kernel(const float* __restrict__ x,
                       const float* __restrict__ bias,
                       const int* __restrict__ pM0,
                       const int* __restrict__ pShift,
                       const int* __restrict__ pZ3,
                       float* __restrict__ out) {
    __shared__ int sBias[NCHAN];
#if USE_ASYNC
    __shared__ f4 sBuf[2][V4_TILE];   // 32 KB double buffer
#endif
    const int tid  = threadIdx.x;
    const int lane = tid & 31;
    const int wave = tid >> 5;

    if (tid < NCHAN) sBias[tid] = __float2int_rn(bias[tid]);
    __syncthreads();

    const int M0    = *pM0;
    const int shift = *pShift;
    const int z3    = *pZ3;
    const int lsh   = shift < 0 ? -shift : 0;
    const int sh    = shift < 0 ? 0 : shift;
    const long long mask = (1ll << sh) - 1ll;
    const long long half = mask >> 1;

    const int blockBase = blockIdx.x * (TILES_PB * V4_TILE);
    const f4* __restrict__ x4 = (const f4*)x;
    f4* __restrict__ o4 = (f4*)out;

#if USE_ASYNC
    auto prefetch = [&](int t, int buf) {
        const int tileBase = blockBase + t * V4_TILE;
#pragma unroll
        for (int k = 0; k < 4; ++k) {
            const int v = wave * V4_WAVE + k * 32 + lane;  // wave-private slice
            const void* gptr = (const void*)(x4 + (tileBase + v));
            void*       lptr = (void*)&sBuf[buf][v];
            __builtin_amdgcn_global_load_async_to_lds_b128(
                (g_v4i_ptr)(void*)(const_cast<void*>(gptr)),
                (l_v4i_ptr)lptr,
                /*offset=*/0, /*cpol=*/0);
        }
    };

    auto compute_tile = [&](int t, int buf) {
        const int tileBase = blockBase + t * V4_TILE;
#pragma unroll
        for (int k = 0; k < 4; ++k) {
            const int v  = wave * V4_WAVE + k * 32 + lane;
            const int gv = tileBase + v;
            const int c  = (int)(((unsigned)gv / HW4) & (NCHAN - 1));
            const int b  = sBias[c];
            f4 xv = sBuf[buf][v];
            f4 o;
            o.x = requant1((int)xv.x + b, lsh, sh, mask, half, M0, z3);
            o.y = requant1((int)xv.y + b, lsh, sh, mask, half, M0, z3);
            o.z = requant1((int)xv.z + b, lsh, sh, mask, half, M0, z3);
            o.w = requant1((int)xv.w + b, lsh, sh, mask, half, M0, z3);
            __builtin_nontemporal_store(o, o4 + gv);   // write-once: TH_NT
        }
    };

    int cur = 0;
    prefetch(0, cur);
#pragma unroll 1
    for (int t = 0; t < TILES_PB - 1; ++t) {
        prefetch(t + 1, cur ^ 1);
        // tile t's 4 async loads are done (done returns in order);
        // tile t+1's 4 may still be outstanding.
        asm volatile("s_wait_asynccnt 4" ::: "memory");
        compute_tile(t, cur);
        cur ^= 1;
    }
    asm volatile("s_wait_asynccnt 0" ::: "memory");
    compute_tile(TILES_PB - 1, cur);
#else
    // Fallback: direct B128 streaming (4 loads in flight per thread per tile).
#pragma unroll 1
    for (int t = 0; t < TILES_PB; ++t) {
        const int tileBase = blockBase + t * V4_TILE;
#pragma unroll
        for (int k = 0; k < 4; ++k) {
            const int v  = wave * V4_WAVE + k * 32 + lane;
            const int gv = tileBase + v;
            const int c  = (int)(((unsigned)gv / HW4) & (NCHAN - 1));
            const int b  = sBias[c];
            f4 xv = x4[gv];
            f4 o;
            o.x = requant1((int)xv.x + b, lsh, sh, mask, half, M0, z3);
            o.y = requant1((int)xv.y + b, lsh, sh, mask, half, M0, z3);
            o.z = requant1((int)xv.z + b, lsh, sh, mask, half, M0, z3);
            o.w = requant1((int)xv.w + b, lsh, sh, mask, half, M0, z3);
            __builtin_nontemporal_store(o, o4 + gv);
        }
    }
#endif
}

extern "C" void kernel_launch(void* const* d_in, const int* in_sizes, int n_in,
                              void* d_out, int out_size, void* d_ws, size_t ws_size,
                              hipStream_t stream) {
    (void)in_sizes; (void)n_in; (void)out_size; (void)d_ws; (void)ws_size;
    const float* x     = (const float*)d_in[0];
    const float* bias  = (const float*)d_in[1];   // [1,256] float
    const int*   pM0   = (const int*)d_in[2];
    const int*   pShf  = (const int*)d_in[3];
    const int*   pZ3   = (const int*)d_in[4];
    float*       out   = (float*)d_out;

    requant_bn_kernel<<<NBLOCKS, THREADS, 0, stream>>>(x, bias, pM0, pShf, pZ3, out);
}